// EfficientSparseDepthwiseConv2d_7112465842616
// MI455X (gfx1250) — compile-verified
//
#include <hip/hip_runtime.h>

// CDNA5 / gfx1250: full-precision fp32 WMMA for the horizontal conv pass.
typedef float v2f __attribute__((ext_vector_type(2)));
typedef float v8f __attribute__((ext_vector_type(8)));

#define KK     7
#define HH     64
#define WW     64
#define CC     256
#define XROWS  22           // 16 output rows + 6 halo
#define XCOLS  72           // 64 cols + 6 halo, padded to 72 (bank stagger)

// Block = 128 threads = 4 waves. One block covers a 16-row x 64-col stripe
// of one (b,c) plane. Each wave owns a 16x16 output tile (w-tile = wave id).
__global__ __launch_bounds__(128)
void dwconv7_wmma_f32(const float* __restrict__ x,
                      const float* __restrict__ wgt,
                      const float* __restrict__ bias,
                      float* __restrict__ out)
{
    __shared__ float Xs[XROWS * XCOLS];       //  6,336 B input halo tile
    __shared__ float Gs[4][XROWS * KK * 16];  // 39,424 B per-wave row-conv results

    const int tid  = threadIdx.x;
    const int bid  = blockIdx.x;
    const int p    = bid >> 2;          // plane index = b*C + c
    const int trow = bid & 3;           // which 16-row stripe
    const int c    = p & (CC - 1);
    const int h0   = trow * 16;
    const long plane_off = (long)p * (HH * WW);

    // ---- cooperative zero-padded halo load: rows h0-3..h0+18, cols -3..68 ----
    for (int idx = tid; idx < XROWS * XCOLS; idx += 128) {
        const int rr = idx / XCOLS;
        const int cc = idx - rr * XCOLS;
        const int gr = h0 - 3 + rr;
        const int gc = cc - 3;
        float v = 0.0f;
        if (gr >= 0 && gr < HH && gc >= 0 && gc < WW)
            v = x[plane_off + gr * WW + gc];
        Xs[idx] = v;
    }

    const int  wv   = tid >> 5;         // wave id -> w tile (0..3)
    const int  lane = tid & 31;
    const int  m    = lane & 15;        // M index (w offset) / N index (kernel row)
    const bool hi   = lane >= 16;
    const int  colb = wv * 16 + m;      // LDS column of tap j=0 for this lane
    const int  j0   = hi ? 2 : 0;       // A taps {0,1} low half, {2,3} high half
    const int  j1   = hi ? 6 : 4;       // second WMMA: taps {4,5} / {6, pad0}

    // ---- B matrix: B[k=j][n=i] = W[i][j]; columns i>=7 are zero ----
    float w0r = 0.f, w1r = 0.f, w2r = 0.f, w3r = 0.f, w4r = 0.f, w5r = 0.f, w6r = 0.f;
    if (m < KK) {
        const float* wp = wgt + c * (KK * KK) + m * KK;   // row i = m of 7x7 kernel
        w0r = wp[0]; w1r = wp[1]; w2r = wp[2]; w3r = wp[3];
        w4r = wp[4]; w5r = wp[5]; w6r = wp[6];
    }
    v2f b0, b1;
    b0.x = hi ? w2r : w0r;   b0.y = hi ? w3r  : w1r;
    b1.x = hi ? w6r : w4r;   b1.y = hi ? 0.0f : w5r;

    __syncthreads();

    // ---- stage 1: horizontal 7-tap conv for 22 input rows via fp32 WMMA ----
    // D[m][n] = sum_j X[r, colbase+m+j] * W[n][j]  (n = kernel row i, 7 of 16 used)
    float* gbase = &Gs[wv][0];
    for (int r = 0; r < XROWS; ++r) {
        const float* xr = &Xs[r * XCOLS + colb];
        v2f a0, a1;
        a0.x = xr[j0];
        a0.y = xr[j0 + 1];
        a1.x = xr[j1];
        // Unconditional load: col <= 48+15+7 = 70 < 72 is always in-bounds;
        // high half-wave (K pad slot, tap 7) is masked to zero with cndmask,
        // avoiding a divergent DS load (saves exec save/restore per row).
        const float t = xr[j1 + 1];
        a1.y = hi ? 0.0f : t;
        v8f acc = {};
        acc = __builtin_amdgcn_wmma_f32_16x16x4_f32(false, a0, false, b0,
                                                    (short)0, acc, false, false);
        acc = __builtin_amdgcn_wmma_f32_16x16x4_f32(false, a1, false, b1,
                                                    (short)0, acc, false, false);
        // D layout: lane holds column n=m, 8 consecutive M (=w) values:
        //   lanes 0-15 -> M 0..7, lanes 16-31 -> M 8..15  => contiguous stores.
        if (m < KK) {
            float* gp = gbase + (r * KK + m) * 16 + (hi ? 8 : 0);
            gp[0] = acc[0]; gp[1] = acc[1]; gp[2] = acc[2]; gp[3] = acc[3];
            gp[4] = acc[4]; gp[5] = acc[5]; gp[6] = acc[6]; gp[7] = acc[7];
        }
    }

    __syncthreads();

    // ---- stage 2: vertical 7-term shifted sum + bias ----
    // out[h][w] = bias + sum_i G[r = h+i][i][w]
    const float bv = bias[c];
    const int  w   = m;
    const int  hb  = hi ? 8 : 0;
    const float* gw = gbase + w;
    for (int hh = 0; hh < 8; ++hh) {
        const int h = hb + hh;
        float acc = bv;
        #pragma unroll
        for (int i = 0; i < KK; ++i)
            acc += gw[((h + i) * KK + i) * 16];
        out[plane_off + (h0 + h) * WW + wv * 16 + w] = acc;
    }
}

extern "C" void kernel_launch(void* const* d_in, const int* in_sizes, int n_in,
                              void* d_out, int out_size, void* d_ws, size_t ws_size,
                              hipStream_t stream) {
    const float* x    = (const float*)d_in[0];   // (16, 256, 64, 64)
    const float* wgt  = (const float*)d_in[1];   // (256, 49)
    const float* bias = (const float*)d_in[2];   // (256,)
    float* out = (float*)d_out;                  // (16, 256, 64, 64)

    const int planes = 16 * CC;                  // B * C
    dim3 grid(planes * 4);                       // 4 row-stripes per plane
    dim3 block(128);
    dwconv7_wmma_f32<<<grid, block, 0, stream>>>(x, wgt, bias, out);
}